// MyModel_87522843558820
// MI455X (gfx1250) — compile-verified
//
#include <hip/hip_runtime.h>
#include <hip/hip_bf16.h>

// CDNA5 (gfx1250) wave32 streaming reduction for:
//   mean(-0.5*(0.4-(z1+z2+3))^2 - 0.5*log(2pi)) = -0.5*mean(t^2) - C,
//   t = z1+z2+2.6
// Roofline: 256 MB read @ 23.3 TB/s ~= 11 us, ~160 MFLOP -> pure HBM-bound.
// Strategy: non-temporal global_load_b128 streams (working set > 192MB L2, so
// NT avoids pollution), f32 accumulate of t^2 only (constants folded into the
// scalar epilogue), v_wmma_f32_16x16x4_f32 wave reduction + ds_swizzle_b32
// finish, per-block partials in d_ws, tiny finalize kernel.

typedef __attribute__((ext_vector_type(2))) float v2f;
typedef __attribute__((ext_vector_type(4))) float v4f;
typedef __attribute__((ext_vector_type(8))) float v8f;

#define HALF_LOG_2PI 0.91893853320467274178f

// One xor-butterfly step via ds_swizzle_b32 (group-of-32 mode):
// offset[15]=0, offset[14:10]=xor_mask, [9:5]=or_mask(0), [4:0]=and_mask(0x1f).
template <int XOR_MASK>
__device__ __forceinline__ float xor_add(float v) {
    int x = __builtin_amdgcn_ds_swizzle(__float_as_int(v),
                                        (XOR_MASK << 10) | 0x1f);
    return v + __int_as_float(x);
}

// Wave-level sum of 32 lane partials using V_WMMA_F32_16X16X4_F32.
// B: VGPR0 = lane partial (rows 0..1 of the 4x16 B), VGPR1 = 0 (rows 2..3).
// A: all ones (16x4).  D[m][n] = acc[n] + acc[n+16], replicated over all rows
// and both lane halves -> dmat[0] in lane l == colsum[l & 15].
// Finish the 16-wide sum with 4 ds_swizzle xor steps; every lane gets total.
__device__ __forceinline__ float wave_reduce_wmma(float acc) {
    v2f a_ones = {1.0f, 1.0f};
    v2f b_vals = {acc, 0.0f};
    v8f c_zero = {};
    v8f dmat = __builtin_amdgcn_wmma_f32_16x16x4_f32(
        /*neg_a=*/false, a_ones, /*neg_b=*/false, b_vals,
        /*c_mod=*/(short)0, c_zero, /*reuse_a=*/false, /*reuse_b=*/false);
    float s = dmat[0];
    s = xor_add<1>(s);
    s = xor_add<2>(s);
    s = xor_add<4>(s);
    s = xor_add<8>(s);
    return s;
}

__global__ void __launch_bounds__(256)
mc_partial_kernel(const v4f* __restrict__ z1, const v4f* __restrict__ z2,
                  float* __restrict__ ws, int n4) {
    const int stride = gridDim.x * blockDim.x;
    float acc = 0.0f;
#pragma unroll 4
    for (int i = blockIdx.x * blockDim.x + threadIdx.x; i < n4; i += stride) {
        // Streamed exactly once: non-temporal (th:TH_LOAD_NT) b128 loads.
        v4f a = __builtin_nontemporal_load(&z1[i]);
        v4f b = __builtin_nontemporal_load(&z2[i]);
        // t = (z1 + z2 + 3) - 0.4 = z1 + z2 + 2.6 ; accumulate t^2
        float t0 = (a.x + b.x) + 2.6f;
        float t1 = (a.y + b.y) + 2.6f;
        float t2 = (a.z + b.z) + 2.6f;
        float t3 = (a.w + b.w) + 2.6f;
        acc = fmaf(t0, t0, acc);
        acc = fmaf(t1, t1, acc);
        acc = fmaf(t2, t2, acc);
        acc = fmaf(t3, t3, acc);
    }

    // EXEC is all-1s here (uniform fallthrough, blockDim multiple of 32).
    float wsum = wave_reduce_wmma(acc);

    __shared__ float lds[8];  // 256 threads / 32 lanes = 8 waves
    const int lane = threadIdx.x & 31;
    const int wave = threadIdx.x >> 5;
    if (lane == 0) lds[wave] = wsum;
    __syncthreads();
    if (threadIdx.x == 0) {
        float s = 0.0f;
        const int nw = blockDim.x >> 5;
        for (int w = 0; w < nw; ++w) s += lds[w];
        ws[blockIdx.x] = s;
    }
}

__global__ void __launch_bounds__(256)
mc_finalize_kernel(const float* __restrict__ ws, int nblocks,
                   float* __restrict__ out, float inv_n) {
    float s = 0.0f;
    for (int i = threadIdx.x; i < nblocks; i += blockDim.x) s += ws[i];
    s = xor_add<16>(s);
    s = xor_add<8>(s);
    s = xor_add<4>(s);
    s = xor_add<2>(s);
    s = xor_add<1>(s);

    __shared__ float lds[8];
    const int lane = threadIdx.x & 31;
    const int wave = threadIdx.x >> 5;
    if (lane == 0) lds[wave] = s;
    __syncthreads();
    if (threadIdx.x == 0) {
        float total = 0.0f;
        const int nw = blockDim.x >> 5;
        for (int w = 0; w < nw; ++w) total += lds[w];
        // mean(log_like) = -0.5 * mean(t^2) - 0.5*log(2*pi)
        out[0] = fmaf(total * inv_n, -0.5f, -HALF_LOG_2PI);
    }
}

extern "C" void kernel_launch(void* const* d_in, const int* in_sizes, int n_in,
                              void* d_out, int out_size, void* d_ws, size_t ws_size,
                              hipStream_t stream) {
    const v4f* z1 = (const v4f*)d_in[0];
    const v4f* z2 = (const v4f*)d_in[1];
    float* out = (float*)d_out;
    float* ws  = (float*)d_ws;

    const int n  = in_sizes[0];       // 33554432
    const int n4 = n >> 2;            // float4 count

    int blocks = 2048;                // 2048 * 256 threads -> 16 float4/thread
    const size_t max_ws = ws_size / sizeof(float);
    if ((size_t)blocks > max_ws) blocks = (int)max_ws;
    if (blocks < 1) blocks = 1;

    mc_partial_kernel<<<blocks, 256, 0, stream>>>(z1, z2, ws, n4);

    const float inv_n = 1.0f / (float)n;  // 2^-25, exact
    mc_finalize_kernel<<<1, 256, 0, stream>>>(ws, blocks, out, inv_n);
}